// GCN_62371515072936
// MI455X (gfx1250) — compile-verified
//
#include <hip/hip_runtime.h>
#include <math.h>

typedef float v2f __attribute__((ext_vector_type(2)));
typedef float v8f __attribute__((ext_vector_type(8)));

__device__ __forceinline__ v8f wmma_f32_k4(v2f a, v2f b, v8f c) {
  // D = A(16x4, f32) x B(4x16, f32) + C(16x16, f32)
  return __builtin_amdgcn_wmma_f32_16x16x4_f32(
      /*neg_a=*/false, a, /*neg_b=*/false, b,
      /*c_mod=*/(short)0, c, /*reuse_a=*/false, /*reuse_b=*/false);
}

// Guaranteed-native fp32 atomic add (no CAS-loop expansion risk).
// Non-returning form: executed at L2 atomic units, tracked with STOREcnt.
__device__ __forceinline__ void atomic_add_f32(float* p, float v) {
  asm volatile("global_atomic_add_f32 %0, %1, off scope:SCOPE_DEV"
               :: "v"(p), "v"(v)
               : "memory");
}

// ---------------- elementwise helpers ----------------

__global__ void k_fill(float* __restrict__ p, float v, int n) {
  int i = blockIdx.x * blockDim.x + threadIdx.x;
  if (i < n) p[i] = v;
}

__global__ void k_deg_accum(const int* __restrict__ dst, float* __restrict__ deg, int E) {
  int i = blockIdx.x * blockDim.x + threadIdx.x;
  if (i < E) atomic_add_f32(&deg[dst[i]], 1.0f);
}

__global__ void k_rsqrt_inplace(float* __restrict__ d, int n) {
  int i = blockIdx.x * blockDim.x + threadIdx.x;
  if (i < n) d[i] = rsqrtf(d[i]);   // deg >= 1 always (self-loop)
}

__global__ void k_norm(const int* __restrict__ src, const int* __restrict__ dst,
                       const float* __restrict__ dinv, float* __restrict__ norm,
                       int E, int N) {
  int i = blockIdx.x * blockDim.x + threadIdx.x;
  if (i >= E + N) return;
  if (i < E) {
    norm[i] = dinv[src[i]] * dinv[dst[i]];
  } else {
    float v = dinv[i - E];
    norm[i] = v * v;
  }
}

__global__ void k_bias_relu(float* __restrict__ a, const float* __restrict__ b, int n) {
  int i = blockIdx.x * blockDim.x + threadIdx.x;
  if (i < n) a[i] = fmaxf(a[i] + b[i & 63], 0.0f);
}

// ---------------- WMMA GEMM: H[M x 64] = X[M x K] * W[K x 64] ----------------
// One wave per 16-row tile; 4 accumulators cover 64 output columns.

__global__ void k_gemm_n64(const float* __restrict__ X, const float* __restrict__ W,
                           float* __restrict__ H, int M, int K) {
  int wave = (int)((blockIdx.x * blockDim.x + threadIdx.x) >> 5);  // uniform per wave
  int lane = threadIdx.x & 31;
  int m0 = wave * 16;
  if (m0 >= M) return;                 // wave-uniform exit -> EXEC all-1s for WMMA
  int half = lane >> 4;                // 0: K pair base {0,1}; 1: base {2,3}
  int lm = lane & 15;

  const float* xrow = X + (size_t)(m0 + lm) * (size_t)K;
  v8f acc0 = {}, acc1 = {}, acc2 = {}, acc3 = {};

  for (int kb = 0; kb < K; kb += 4) {
    int k0 = kb + 2 * half;
    v2f a;
    a.x = xrow[k0];
    a.y = xrow[k0 + 1];
    const float* w0 = W + (size_t)k0 * 64;   // row k0 ; row k0+1 = w0+64
    v2f b0, b1, b2, b3;
    b0.x = w0[lm];       b0.y = w0[64 + lm];
    b1.x = w0[16 + lm];  b1.y = w0[64 + 16 + lm];
    b2.x = w0[32 + lm];  b2.y = w0[64 + 32 + lm];
    b3.x = w0[48 + lm];  b3.y = w0[64 + 48 + lm];
    acc0 = wmma_f32_k4(a, b0, acc0);
    acc1 = wmma_f32_k4(a, b1, acc1);
    acc2 = wmma_f32_k4(a, b2, acc2);
    acc3 = wmma_f32_k4(a, b3, acc3);
  }

#pragma unroll
  for (int r = 0; r < 8; ++r) {
    int row = m0 + r + 8 * half;
    float* hr = H + (size_t)row * 64 + lm;
    hr[0]  = acc0[r];
    hr[16] = acc1[r];
    hr[32] = acc2[r];
    hr[48] = acc3[r];
  }
}

// ---------------- edge scatter: agg[dst] += h[src] * norm ----------------
// 16 threads per edge; each handles 4 channels (float4 gather + 4 native f32 atomics).

__global__ void k_scatter(const int* __restrict__ src, const int* __restrict__ dst,
                          const float* __restrict__ norm, const float* __restrict__ h,
                          float* __restrict__ agg, int E, int N) {
  unsigned int gid = blockIdx.x * blockDim.x + threadIdx.x;
  unsigned int eid = gid >> 4;
  int t4 = (gid & 15) * 4;
  if (eid >= (unsigned int)(E + N)) return;
  int s, d;
  if (eid < (unsigned int)E) {
    s = src[eid];
    d = dst[eid];
  } else {
    s = d = (int)eid - E;
  }
  float nm = norm[eid];
  const float4 hv = *reinterpret_cast<const float4*>(h + (size_t)s * 64 + t4);
  float* ap = agg + (size_t)d * 64 + t4;
  atomic_add_f32(ap + 0, hv.x * nm);
  atomic_add_f32(ap + 1, hv.y * nm);
  atomic_add_f32(ap + 2, hv.z * nm);
  atomic_add_f32(ap + 3, hv.w * nm);
}

// ---------------- global mean pool (accumulate) ----------------

__global__ void k_pool(const float* __restrict__ h, const int* __restrict__ batch,
                       float* __restrict__ gsum, float* __restrict__ cnt, int N) {
  unsigned int gid = blockIdx.x * blockDim.x + threadIdx.x;
  if (gid >= (unsigned int)N * 64u) return;
  int c = gid & 63;
  int n = gid >> 6;
  int g = batch[n];
  atomic_add_f32(&gsum[g * 64 + c], h[(size_t)n * 64 + c]);
  if (c == 0) atomic_add_f32(&cnt[g], 1.0f);
}

// ---------------- final: out = (gsum/max(cnt,1)) @ Wp + bp ----------------
// One block, 16 waves; wave w does the 16x16 tile (w/4, w%4).

__global__ void k_pool_gemm_out(const float* __restrict__ gsum, const float* __restrict__ cnt,
                                const float* __restrict__ Wp, const float* __restrict__ bp,
                                float* __restrict__ out) {
  int wave = threadIdx.x >> 5;
  int lane = threadIdx.x & 31;
  int mi = wave >> 2, ni = wave & 3;
  int half = lane >> 4, lm = lane & 15;

  int arow = mi * 16 + lm;
  float inv = 1.0f / fmaxf(cnt[arow], 1.0f);

  v8f acc = {};
  for (int kb = 0; kb < 64; kb += 4) {
    int k0 = kb + 2 * half;
    v2f a;
    a.x = gsum[arow * 64 + k0] * inv;
    a.y = gsum[arow * 64 + k0 + 1] * inv;
    v2f b;
    b.x = Wp[k0 * 64 + ni * 16 + lm];
    b.y = Wp[(k0 + 1) * 64 + ni * 16 + lm];
    acc = wmma_f32_k4(a, b, acc);
  }

#pragma unroll
  for (int r = 0; r < 8; ++r) {
    int row = mi * 16 + r + 8 * half;
    out[row * 64 + ni * 16 + lm] = acc[r] + bp[ni * 16 + lm];
  }
}

// ---------------- launch ----------------

static inline size_t align_f(size_t n) { return (n + 1023) & ~(size_t)1023; }

extern "C" void kernel_launch(void* const* d_in, const int* in_sizes, int n_in,
                              void* d_out, int out_size, void* d_ws, size_t ws_size,
                              hipStream_t stream) {
  const float* x  = (const float*)d_in[0];
  const float* W1 = (const float*)d_in[1];
  const float* b1 = (const float*)d_in[2];
  const float* W2 = (const float*)d_in[3];
  const float* b2 = (const float*)d_in[4];
  const float* Wp = (const float*)d_in[5];
  const float* bp = (const float*)d_in[6];
  const int* ei   = (const int*)d_in[7];
  const int* batch = (const int*)d_in[8];

  const int E = in_sizes[7] / 2;
  const int N = in_sizes[8];
  const int K1 = in_sizes[0] / N;   // 128
  const int* src = ei;
  const int* dst = ei + E;

  float* ws = (float*)d_ws;
  size_t o = 0;
  float* dinv = ws + o;  o += align_f((size_t)N);
  float* norm = ws + o;  o += align_f((size_t)(E + N));
  float* h    = ws + o;  o += align_f((size_t)N * 64);
  float* agg  = ws + o;  o += align_f((size_t)N * 64);
  float* gsum = ws + o;  o += 64 * 64;
  float* cnt  = ws + o;  o += 64;
  (void)ws_size;

  const int T = 256;
  auto blocks = [](long long n, int t) { return (unsigned int)((n + t - 1) / t); };

  // 1) degree (self-loop counts as 1) -> dinv -> norm
  k_fill<<<blocks(N, T), T, 0, stream>>>(dinv, 1.0f, N);
  k_deg_accum<<<blocks(E, T), T, 0, stream>>>(dst, dinv, E);
  k_rsqrt_inplace<<<blocks(N, T), T, 0, stream>>>(dinv, N);
  k_norm<<<blocks((long long)E + N, T), T, 0, stream>>>(src, dst, dinv, norm, E, N);

  const int tiles = (N + 15) / 16;
  const unsigned int gemm_blocks = (unsigned int)((tiles + 7) / 8);  // 8 waves/block
  const long long NE16 = (long long)(E + N) * 16;

  // 2) layer 1: h = x@W1 ; agg = scatter(h) ; relu(agg + b1)
  k_gemm_n64<<<gemm_blocks, T, 0, stream>>>(x, W1, h, N, K1);
  k_fill<<<blocks((long long)N * 64, T), T, 0, stream>>>(agg, 0.0f, N * 64);
  k_scatter<<<blocks(NE16, T), T, 0, stream>>>(src, dst, norm, h, agg, E, N);
  k_bias_relu<<<blocks((long long)N * 64, T), T, 0, stream>>>(agg, b1, N * 64);

  // 3) layer 2: h = agg@W2 ; agg = scatter(h) ; relu(agg + b2)
  k_gemm_n64<<<gemm_blocks, T, 0, stream>>>(agg, W2, h, N, 64);
  k_fill<<<blocks((long long)N * 64, T), T, 0, stream>>>(agg, 0.0f, N * 64);
  k_scatter<<<blocks(NE16, T), T, 0, stream>>>(src, dst, norm, h, agg, E, N);
  k_bias_relu<<<blocks((long long)N * 64, T), T, 0, stream>>>(agg, b2, N * 64);

  // 4) mean pool + output projection
  k_fill<<<blocks(64 * 64 + 64, T), T, 0, stream>>>(gsum, 0.0f, 64 * 64 + 64); // gsum then cnt
  k_pool<<<blocks((long long)N * 64, T), T, 0, stream>>>(agg, batch, gsum, cnt, N);
  k_pool_gemm_out<<<1, 512, 0, stream>>>(gsum, cnt, Wp, bp, (float*)d_out);
}